// MultiModalCLGAE_54743653154835
// MI455X (gfx1250) — compile-verified
//
#include <hip/hip_runtime.h>
#include <hip/hip_bf16.h>

typedef __attribute__((ext_vector_type(2))) float v2f;
typedef __attribute__((ext_vector_type(8))) float v8f;

static __device__ __forceinline__ void atomic_add_f32(float* p, float v) {
  // native GLOBAL_ATOMIC_ADD_F32 (no CAS loop)
  unsafeAtomicAdd(p, v);
}

static inline int cdiv(long long a, long long b) { return (int)((a + b - 1) / b); }

// ---------------- degree / norm ----------------
__global__ void k_init_deg(float* deg, int N) {
  int i = blockIdx.x * blockDim.x + threadIdx.x;
  if (i < N) deg[i] = 1.0f;  // self-loop contribution
}

__global__ void k_count_deg(const int* __restrict__ edst, float* deg, int E) {
  int e = blockIdx.x * blockDim.x + threadIdx.x;
  if (e < E) atomic_add_f32(&deg[edst[e]], 1.0f);
}

__global__ void k_dinv(float* deg, int N) {
  int i = blockIdx.x * blockDim.x + threadIdx.x;
  if (i < N) deg[i] = rsqrtf(deg[i]);  // deg >= 1 always (self-loops)
}

// ---------------- GEMM via V_WMMA_F32_16X16X4_F32 ----------------
// C[M x (NT*16)] = op(A[M x 128]) * W[128 x (NT*16)],  op = identity or relu
template <int NT, bool RELU>
__global__ void k_gemm_wmma(const float* __restrict__ A, const float* __restrict__ W,
                            float* __restrict__ C, int M) {
  const int N = NT * 16;
  const int lane = threadIdx.x & 31;
  const int wave = blockIdx.x * (blockDim.x >> 5) + (threadIdx.x >> 5);
  const int mtile = wave;
  if (mtile * 16 >= M) return;               // whole-wave exit: EXEC stays all-1s for WMMA
  const int hf  = lane >> 4;                  // 0: lanes 0-15, 1: lanes 16-31
  const int l16 = lane & 15;

  int arow = mtile * 16 + l16;
  if (arow > M - 1) arow = M - 1;             // M % 16 == 0 for this problem; safety clamp
  const float* __restrict__ ap = A + (size_t)arow * 128;

  v8f acc[NT] = {};
  #pragma unroll 4
  for (int kt = 0; kt < 32; ++kt) {
    const int k = kt * 4 + 2 * hf;            // this lane's K pair (A and B share K layout)
    float a0 = ap[k], a1 = ap[k + 1];
    if (RELU) { a0 = fmaxf(a0, 0.0f); a1 = fmaxf(a1, 0.0f); }
    v2f a; a.x = a0; a.y = a1;
    #pragma unroll
    for (int nt = 0; nt < NT; ++nt) {
      const int col = nt * 16 + l16;
      v2f b; b.x = W[(size_t)k * N + col]; b.y = W[(size_t)(k + 1) * N + col];
      acc[nt] = __builtin_amdgcn_wmma_f32_16x16x4_f32(
          /*neg_a=*/false, a, /*neg_b=*/false, b,
          /*c_mod=*/(short)0, acc[nt], /*reuse_a=*/false, /*reuse_b=*/false);
    }
  }
  // C/D layout: VGPR j -> row (j + 8*hf), col l16
  #pragma unroll
  for (int nt = 0; nt < NT; ++nt) {
    const int col = nt * 16 + l16;
    #pragma unroll
    for (int j = 0; j < 8; ++j) {
      const int r = mtile * 16 + j + 8 * hf;
      if (r < M) C[(size_t)r * N + col] = acc[nt][j];
    }
  }
}

// ---------------- self-loop + bias init: agg = hw * dinv^2 + b ----------------
template <int C>
__global__ void k_self_bias(const float* __restrict__ hw, const float* __restrict__ dinv,
                            const float* __restrict__ bias, float* __restrict__ agg, int N) {
  long long t = (long long)blockIdx.x * blockDim.x + threadIdx.x;
  if (t >= (long long)N * C) return;
  const int i = (int)(t / C), c = (int)(t % C);
  const float di = dinv[i];
  agg[t] = hw[t] * di * di + bias[c];
}

// ---------------- edge scatter: agg[dst] += hw[src] * dinv[src]*dinv[dst] ----------------
template <int C>
__global__ void k_edge_scatter(const float* __restrict__ hw, const int* __restrict__ esrc,
                               const int* __restrict__ edst, const float* __restrict__ dinv,
                               float* __restrict__ agg, int E) {
  const int lane = threadIdx.x & 31;
  const int e = blockIdx.x * (blockDim.x >> 5) + (threadIdx.x >> 5);
  if (e >= E) return;
  const int s = esrc[e], d = edst[e];          // uniform per wave -> broadcast loads
  const float w = dinv[s] * dinv[d];
  constexpr int V = C / 32;                    // floats per lane (4 for C=128, 2 for C=64)
  const float* __restrict__ sp = hw + (size_t)s * C + lane * V;
  float* __restrict__ dp = agg + (size_t)d * C + lane * V;
  float v[V];
  #pragma unroll
  for (int j = 0; j < V; ++j) v[j] = sp[j];    // coalesced B128/B64 gather per wave
  #pragma unroll
  for (int j = 0; j < V; ++j) atomic_add_f32(dp + j, v[j] * w);
}

// ---------------- pooling ----------------
__global__ void k_zero(float* p, int n) {
  int i = blockIdx.x * blockDim.x + threadIdx.x;
  if (i < n) p[i] = 0.0f;
}

__global__ void k_count_batch(const int* __restrict__ batch, float* cnt, int N) {
  int i = blockIdx.x * blockDim.x + threadIdx.x;
  if (i < N) atomic_add_f32(&cnt[batch[i]], 1.0f);
}

__global__ void k_pool_accum(const float* __restrict__ z, const int* __restrict__ batch,
                             float* pool, int N) {
  long long t = (long long)blockIdx.x * blockDim.x + threadIdx.x;
  if (t >= (long long)N * 64) return;
  const int i = (int)(t >> 6), c = (int)(t & 63);
  atomic_add_f32(&pool[(size_t)batch[i] * 64 + c], z[t]);
}

__global__ void k_finalize(const float* __restrict__ pool, const float* __restrict__ cnt,
                           float* __restrict__ out, int G) {
  int t = blockIdx.x * blockDim.x + threadIdx.x;
  if (t >= G * 64) return;
  out[t] = pool[t] / fmaxf(cnt[t >> 6], 1.0f);
}

extern "C" void kernel_launch(void* const* d_in, const int* in_sizes, int n_in,
                              void* d_out, int out_size, void* d_ws, size_t ws_size,
                              hipStream_t stream) {
  const float* x   = (const float*)d_in[0];
  const float* W1  = (const float*)d_in[1];
  const float* b1  = (const float*)d_in[2];
  const float* W2  = (const float*)d_in[3];
  const float* b2  = (const float*)d_in[4];
  const int* eidx  = (const int*)d_in[5];
  const int* batch = (const int*)d_in[6];

  const int N = in_sizes[0] / 128;
  const int E = in_sizes[5] / 2;
  const int G = out_size / 64;
  const int* esrc = eidx;
  const int* edst = eidx + E;

  // workspace carve-out (peak ~52 MB): dinv | bufA (hw) | bufB (agg) | pool | cnt
  char* ws = (char*)d_ws;
  size_t off = 0;
  auto carve = [&](size_t bytes) {
    size_t o = off;
    off = (off + bytes + 255) & ~(size_t)255;
    return o;
  };
  float* dinv = (float*)(ws + carve((size_t)N * 4));
  float* bufA = (float*)(ws + carve((size_t)N * 128 * 4));
  float* bufB = (float*)(ws + carve((size_t)N * 128 * 4));
  float* pool = (float*)(ws + carve((size_t)G * 64 * 4));
  float* cnt  = (float*)(ws + carve((size_t)G * 4));
  if (off > ws_size) return;  // workspace insufficient (not expected)

  const int T = 256;
  // degrees -> dinv (in place)
  k_init_deg<<<cdiv(N, T), T, 0, stream>>>(dinv, N);
  k_count_deg<<<cdiv(E, T), T, 0, stream>>>(edst, dinv, E);
  k_dinv<<<cdiv(N, T), T, 0, stream>>>(dinv, N);

  const int mtiles = cdiv(N, 16);
  const int gemm_blocks = cdiv(mtiles, 4);  // 4 waves (128 threads) per block

  // layer 1: hw1 = x @ W1 ; agg1 = scatter(hw1) + self + b1
  k_gemm_wmma<8, false><<<gemm_blocks, 128, 0, stream>>>(x, W1, bufA, N);
  k_self_bias<128><<<cdiv((long long)N * 128, T), T, 0, stream>>>(bufA, dinv, b1, bufB, N);
  k_edge_scatter<128><<<cdiv(E, 8), T, 0, stream>>>(bufA, esrc, edst, dinv, bufB, E);

  // layer 2: hw2 = relu(agg1) @ W2 ; agg2 = scatter(hw2) + self + b2
  k_gemm_wmma<4, true><<<gemm_blocks, 128, 0, stream>>>(bufB, W2, bufA, N);
  k_self_bias<64><<<cdiv((long long)N * 64, T), T, 0, stream>>>(bufA, dinv, b2, bufB, N);
  k_edge_scatter<64><<<cdiv(E, 8), T, 0, stream>>>(bufA, esrc, edst, dinv, bufB, E);

  // global mean pool
  k_zero<<<cdiv(G * 64, T), T, 0, stream>>>(pool, G * 64);
  k_zero<<<cdiv(G, T), T, 0, stream>>>(cnt, G);
  k_count_batch<<<cdiv(N, T), T, 0, stream>>>(batch, cnt, N);
  k_pool_accum<<<cdiv((long long)N * 64, T), T, 0, stream>>>(bufB, batch, pool, N);
  k_finalize<<<cdiv(G * 64, T), T, 0, stream>>>(pool, cnt, (float*)d_out, G);
}